// SpikingTransformerNetwork_28810640622385
// MI455X (gfx1250) — compile-verified
//
#include <hip/hip_runtime.h>

// ---------------- problem constants (from reference) ----------------
#define IN_D   544
#define HID_D  1024
#define OUT_D  2048

#define REST    (-70.0f)
#define RESET   (-65.0f)
#define THRESH  (-55.0f)
#define ALPHA   0.95f
#define BETA    0.8f
#define H_TH    1.0f
#define NU1_PRE  0.001f
#define NU1_POST 0.01f
#define NU2      0.0001f
#define GAIN     20.0f

// exp(-dt/tau) constants
#define DECAY_V  0.904837418035960f   // exp(-1/10)
#define DECAY_TR 0.951229424500714f   // exp(-1/20)
#define DECAY_PM 0.951229424500714f   // exp(-1/20)

typedef float v2f __attribute__((ext_vector_type(2)));
typedef float v8f __attribute__((ext_vector_type(8)));

// ---------------- input LIF ----------------
__global__ void input_lif_kernel(const float* __restrict__ x,
                                 float* __restrict__ v_in,
                                 float* __restrict__ tr_in,
                                 float* __restrict__ s_in) {
  int i = blockIdx.x * blockDim.x + threadIdx.x;
  if (i >= IN_D) return;
  float v = DECAY_V * (v_in[i] - REST) + REST + x[i] * GAIN;
  float s = (v >= THRESH) ? 1.0f : 0.0f;
  v_in[i]  = (s > 0.0f) ? RESET : v;
  tr_in[i] = (s > 0.0f) ? 1.0f  : tr_in[i] * DECAY_TR;
  s_in[i]  = s;
}

// ---------------- hidden: i_hid = s_in @ w1 ; RUM dynamics ----------------
// 256 threads = 4 reduction segments x 64 output columns; fixed-order LDS
// combine (deterministic). Spike value is wave-uniform -> uniform row skip.
__global__ void hidden_kernel(const float* __restrict__ s_in,
                              const float* __restrict__ w1,
                              float* __restrict__ v_hid,
                              float* __restrict__ b_hid,
                              float* __restrict__ tr_hid,
                              float* __restrict__ p_plus,
                              float* __restrict__ s_hid) {
  __shared__ float ls[IN_D];
  __shared__ float red[256];
  int tid = threadIdx.x;
  for (int i = tid; i < IN_D; i += 256) ls[i] = s_in[i];
  __syncthreads();

  int seg  = tid >> 6;                    // 0..3 (uniform per wave)
  int hoff = tid & 63;
  int h    = blockIdx.x * 64 + hoff;      // grid = HID_D/64 = 16 blocks
  const int seg_len = (IN_D + 3) / 4;     // 136
  int i0 = seg * seg_len;
  int i1 = (i0 + seg_len < IN_D) ? i0 + seg_len : IN_D;

  float acc = 0.0f;
  for (int i = i0; i < i1; ++i) {
    if (ls[i] != 0.0f) acc += w1[i * HID_D + h];
  }
  red[tid] = acc;
  __syncthreads();

  if (seg == 0) {                         // 2 full waves, uniform branch
    float isum = red[hoff] + red[hoff + 64] + red[hoff + 128] + red[hoff + 192];
    float v = ALPHA * v_hid[h] + isum;
    float b = b_hid[h];
    float s = (v >= H_TH + b) ? 1.0f : 0.0f;
    v_hid[h]  = (s > 0.0f) ? 0.0f : v;
    b_hid[h]  = BETA * b + s;
    tr_hid[h] = (s > 0.0f) ? 1.0f : tr_hid[h] * DECAY_TR;
    p_plus[h] = DECAY_PM * p_plus[h] + s;     // A_PLUS = 1
    s_hid[h]  = s;
  }
}

// ---------------- output: i_out = s_hid @ w2 ; LIF ; raster row ----------------
__global__ void output_kernel(const float* __restrict__ s_hid,
                              const float* __restrict__ w2,
                              float* __restrict__ v_out,
                              float* __restrict__ p_minus,
                              float* __restrict__ out_row) {
  __shared__ float ls[HID_D];
  __shared__ float red[256];
  int tid = threadIdx.x;
  for (int i = tid; i < HID_D; i += 256) ls[i] = s_hid[i];
  __syncthreads();

  int seg  = tid >> 6;                    // 0..3
  int ooff = tid & 63;
  int o    = blockIdx.x * 64 + ooff;      // grid = OUT_D/64 = 32 blocks
  const int seg_len = HID_D / 4;          // 256
  int i0 = seg * seg_len;
  int i1 = i0 + seg_len;

  float acc = 0.0f;
  for (int i = i0; i < i1; ++i) {
    if (ls[i] != 0.0f) acc += w2[i * OUT_D + o];
  }
  red[tid] = acc;
  __syncthreads();

  if (seg == 0) {
    float isum = red[ooff] + red[ooff + 64] + red[ooff + 128] + red[ooff + 192];
    float v = DECAY_V * (v_out[o] - REST) + REST + isum;
    float s = (v >= THRESH) ? 1.0f : 0.0f;
    v_out[o]   = (s > 0.0f) ? RESET : v;
    p_minus[o] = DECAY_PM * p_minus[o] - s;   // A_MINUS = -1
    out_row[o] = s;
  }
}

// ---------------- WMMA rank-2 tile update ----------------
// One wave, one 16x16 tile:  D = A(16x4) x B(4x16) + C
//   A cols: [sa*f1 | sb*f2 | sa*f1 | sb*f2],  B rows: [g1 | g2 | 0 | 0]
// => W_tile += sa*(f1 ⊗ g1) + sb*(f2 ⊗ g2) in one v_wmma_f32_16x16x4_f32.
// A factors replicated into both lane-halves; B upper lane-half zeroed, so the
// result is invariant to the K<->lane-half mapping of the B fragment.
template <bool CLAMP01>
__device__ __forceinline__ void tile_rank2_update(
    float* __restrict__ W, int ncols, int r0, int c0,
    const float* __restrict__ f1, float sa,
    const float* __restrict__ f2, float sb,
    const float* __restrict__ g1, const float* __restrict__ g2, int lane) {
  int  mh = lane & 15;
  bool lo = lane < 16;

  v2f a, b;
  a.x = sa * f1[r0 + mh];
  a.y = sb * f2[r0 + mh];
  b.x = lo ? g1[c0 + mh] : 0.0f;
  b.y = lo ? g2[c0 + mh] : 0.0f;

  // C/D fragment: VGPR r, lanes 0-15 -> (M=r, N=lane); lanes 16-31 -> (M=r+8)
  int Mb = lo ? 0 : 8;
  v8f c;
#pragma unroll
  for (int r = 0; r < 8; ++r)
    c[r] = W[(r0 + Mb + r) * ncols + c0 + mh];

  c = __builtin_amdgcn_wmma_f32_16x16x4_f32(false, a, false, b, (short)0, c,
                                            false, false);

#pragma unroll
  for (int r = 0; r < 8; ++r) {
    float v = c[r];
    if (CLAMP01) v = fminf(fmaxf(v, 0.0f), 1.0f);
    W[(r0 + Mb + r) * ncols + c0 + mh] = v;
  }
}

#define W1_TILES ((IN_D / 16) * (HID_D / 16))   // 2176 (= 272 blocks * 8 waves)
#define W2_TILES ((HID_D / 16) * (OUT_D / 16))  // 8192

// Fused STDP update for both weight matrices (one launch per step).
// Branches are wave-uniform (tile id uniform per wave) -> EXEC all-ones at WMMA.
__global__ void weight_update_kernel(float* __restrict__ w1,
                                     float* __restrict__ w2,
                                     const float* __restrict__ tr_in,
                                     const float* __restrict__ s_in,
                                     const float* __restrict__ s_hid,
                                     const float* __restrict__ tr_hid,
                                     const float* __restrict__ p_plus,
                                     const float* __restrict__ s_out,
                                     const float* __restrict__ p_minus) {
  int wave = (blockIdx.x * blockDim.x + threadIdx.x) >> 5;  // 8 waves / block
  int lane = threadIdx.x & 31;
  if (wave < W1_TILES) {
    // w1 = clip(w1 + NU1_POST*tr_in⊗s_hid - NU1_PRE*s_in⊗tr_hid, 0, 1)
    const int tcols = HID_D / 16;
    int tr = wave / tcols, tc = wave % tcols;
    tile_rank2_update<true>(w1, HID_D, tr << 4, tc << 4,
                            tr_in, NU1_POST, s_in, -NU1_PRE, s_hid, tr_hid, lane);
  } else {
    // w2 += NU2*(p_plus⊗s_out + s_hid⊗p_minus)
    int w = wave - W1_TILES;
    const int tcols = OUT_D / 16;
    int tr = w / tcols, tc = w % tcols;
    tile_rank2_update<false>(w2, OUT_D, tr << 4, tc << 4,
                             p_plus, NU2, s_hid, NU2, s_out, p_minus, lane);
  }
}

// ---------------- state init ----------------
__global__ void init_state_kernel(float* v_in, float* v_hid, float* v_out,
                                  float* tr_in, float* tr_hid, float* b_hid,
                                  float* p_plus, float* p_minus) {
  int i = blockIdx.x * blockDim.x + threadIdx.x;
  if (i < IN_D)  { v_in[i] = REST; tr_in[i] = 0.0f; }
  if (i < HID_D) { v_hid[i] = 0.0f; tr_hid[i] = 0.0f; b_hid[i] = 0.0f; p_plus[i] = 0.0f; }
  if (i < OUT_D) { v_out[i] = REST; p_minus[i] = 0.0f; }
}

// ---------------- host launcher ----------------
extern "C" void kernel_launch(void* const* d_in, const int* in_sizes, int n_in,
                              void* d_out, int out_size, void* d_ws, size_t ws_size,
                              hipStream_t stream) {
  const float* x     = (const float*)d_in[0];
  const float* w1_in = (const float*)d_in[1];
  const float* w2_in = (const float*)d_in[2];
  float* out = (float*)d_out;
  int T = out_size / OUT_D;

  size_t need = (size_t)IN_D * HID_D + (size_t)HID_D * OUT_D +
                3 * (size_t)IN_D + 5 * (size_t)HID_D + 2 * (size_t)OUT_D + 64;
  if (ws_size < need * sizeof(float)) return;

  float* p  = (float*)d_ws;
  float* w1      = p; p += (size_t)IN_D * HID_D;   // mutable copies, L2-resident
  float* w2      = p; p += (size_t)HID_D * OUT_D;
  float* v_in    = p; p += IN_D;
  float* tr_in   = p; p += IN_D;
  float* s_in    = p; p += IN_D;
  float* v_hid   = p; p += HID_D;
  float* tr_hid  = p; p += HID_D;
  float* b_hid   = p; p += HID_D;
  float* p_plus  = p; p += HID_D;
  float* s_hid   = p; p += HID_D;
  float* v_out   = p; p += OUT_D;
  float* p_minus = p; p += OUT_D;

  hipMemcpyAsync(w1, w1_in, (size_t)IN_D * HID_D * sizeof(float),
                 hipMemcpyDeviceToDevice, stream);
  hipMemcpyAsync(w2, w2_in, (size_t)HID_D * OUT_D * sizeof(float),
                 hipMemcpyDeviceToDevice, stream);
  init_state_kernel<<<(OUT_D + 255) / 256, 256, 0, stream>>>(
      v_in, v_hid, v_out, tr_in, tr_hid, b_hid, p_plus, p_minus);

  const int UPD_BLOCKS = (W1_TILES + W2_TILES) / 8;  // 1296

  for (int t = 0; t < T; ++t) {
    float* row = out + (size_t)t * OUT_D;
    input_lif_kernel<<<(IN_D + 255) / 256, 256, 0, stream>>>(x, v_in, tr_in, s_in);
    hidden_kernel<<<HID_D / 64, 256, 0, stream>>>(s_in, w1, v_hid, b_hid,
                                                  tr_hid, p_plus, s_hid);
    output_kernel<<<OUT_D / 64, 256, 0, stream>>>(s_hid, w2, v_out, p_minus, row);
    weight_update_kernel<<<UPD_BLOCKS, 256, 0, stream>>>(
        w1, w2, tr_in, s_in, s_hid, tr_hid, p_plus, row, p_minus);
  }
}